// CustomLoss_11630771438153
// MI455X (gfx1250) — compile-verified
//
#include <hip/hip_runtime.h>
#include <math.h>
#include <stdint.h>

#define KNN      16
#define TAU      0.1f
#define BETA     1.0f
#define LAMB     1e-4f
#define EPSF     1e-8f
#define DIM      128
#define CHUNK    2048
#define STEP     128
#define XPAD     132            // 128 + 4 DW padding (TDM pad_interval=128DW, pad_amount=4DW)

typedef float v2f __attribute__((ext_vector_type(2)));
typedef float v8f __attribute__((ext_vector_type(8)));
typedef unsigned int u32x4 __attribute__((ext_vector_type(4)));
typedef int i32x4 __attribute__((ext_vector_type(4)));
typedef int i32x8 __attribute__((ext_vector_type(8)));

// ---- Tensor Data Mover: DMA one 128xDIM f32 tile (row-major) into LDS ----
// LDS destination uses stride XPAD floats via hardware padding:
//   pad_interval code 6 -> 2^(6+1)=128 DWORDs, pad_amount code 3 -> 4 DWORDs.
__device__ __forceinline__ void tdm_load_tile(uint32_t lds_off, const float* gsrc,
                                              int rows_avail) {
    unsigned long long ga = (unsigned long long)(uintptr_t)gsrc;
    u32x4 g0;
    g0.x = 1u;                                           // count=1, user mode
    g0.y = lds_off;                                      // lds_addr
    g0.z = (unsigned)(ga & 0xffffffffull);               // global_addr[31:0]
    g0.w = (unsigned)((ga >> 32) & 0x01ffffffull) | (2u << 30);  // addr[56:32] | type=2
    unsigned rows = (unsigned)rows_avail;
    i32x8 g1;
    g1[0] = (int)((2u << 16) | (1u << 20) | (6u << 22) | (3u << 25)); // data_size=4B, pad en/int/amt
    g1[1] = (int)(128u << 16);                           // tensor_dim0 lo16 (=128) in [31:16]
    g1[2] = (int)((rows & 0xffffu) << 16);               // dim0 hi16=0 | tensor_dim1 lo16
    g1[3] = (int)(((rows >> 16) & 0xffffu) | (128u << 16)); // dim1 hi16 | tile_dim0=128
    g1[4] = (int)128;                                    // tile_dim1=128, tile_dim2=0
    g1[5] = (int)128;                                    // tensor_dim0_stride lo32 = 128
    g1[6] = 0;                                           // stride0 hi | stride1 lo
    g1[7] = 0;
    i32x4 z4 = {0, 0, 0, 0};
#if __clang_major__ >= 23
    i32x8 z8 = {0, 0, 0, 0, 0, 0, 0, 0};
    __builtin_amdgcn_tensor_load_to_lds(g0, g1, z4, z4, z8, 0);
#else
    __builtin_amdgcn_tensor_load_to_lds(g0, g1, z4, z4, 0);
#endif
}

// ---------------- Kernel 1: Tq = q_batch @ W  (fp32, tiny) ----------------
__global__ __launch_bounds__(128) void tq_project(const float* __restrict__ q,
                                                  const float* __restrict__ W,
                                                  float* __restrict__ Tq) {
    int b = blockIdx.x, c = threadIdx.x;
    float s = 0.f;
    #pragma unroll 8
    for (int k = 0; k < DIM; ++k) s = fmaf(q[b * DIM + k], W[k * DIM + c], s);
    Tq[b * DIM + c] = s;
}

// ---------------- Kernel 1b: ||x||^2 for every database row ---------------
__global__ __launch_bounds__(256) void x_norms(const float* __restrict__ X,
                                               float* __restrict__ xnorm, int N) {
    int i = blockIdx.x * 256 + threadIdx.x;
    if (i < N) {
        const float4* xr = (const float4*)(X + (size_t)i * DIM);
        float s = 0.f;
        #pragma unroll
        for (int k = 0; k < DIM / 4; ++k) {
            float4 v = xr[k];
            s = fmaf(v.x, v.x, s); s = fmaf(v.y, v.y, s);
            s = fmaf(v.z, v.z, s); s = fmaf(v.w, v.w, s);
        }
        xnorm[i] = s;
    }
}

// ---- Kernel 2: TDM double-buffered f32-WMMA distances + chunk top-16 -----
// grid.x = B/16 row-groups, grid.y = chunks of CHUNK database points.
__global__ __launch_bounds__(256) void knn_dist_topk(const float* __restrict__ Tq,
                                                     const float* __restrict__ X,
                                                     const float* __restrict__ xnorm,
                                                     int N, int nch,
                                                     float* __restrict__ candD,
                                                     int* __restrict__ candI) {
    const int rg = blockIdx.x;
    const int ch = blockIdx.y;
    const int t = threadIdx.x;
    const int lane = t & 31, wave = t >> 5;

    __shared__ float tqs[16 * XPAD];
    __shared__ float xbuf[2][STEP * XPAD];   // double-buffered TDM destination
    __shared__ float dist[16 * XPAD];

    const int cbase = ch * CHUNK;
    const int csize = (cbase + CHUNK > N) ? (N - cbase) : CHUNK;
    const int nsteps = (csize + STEP - 1) / STEP;

    // prefetch step 0 via TDM (one wave issues; TENSORcnt tracks it)
    if (wave == 0)
        tdm_load_tile((uint32_t)(uintptr_t)&xbuf[0][0], X + (size_t)cbase * DIM, N - cbase);

    // stage Tq tile
    for (int i = t; i < 16 * DIM; i += 256) {
        int r = i >> 7, c = i & 127;
        tqs[r * XPAD + c] = Tq[(rg * 16 + r) * DIM + c];
    }
    __syncthreads();

    // hoist A-fragments (16x4 f32; lanes 0-15: K={0,1}, lanes 16-31: K={2,3})
    v2f afrag[32];
    {
        const int m = lane & 15;
        const int koff = (lane < 16) ? 0 : 2;
        #pragma unroll
        for (int kb = 0; kb < 32; ++kb)
            afrag[kb] = *(const v2f*)&tqs[m * XPAD + kb * 4 + koff];
    }

    // per-thread running top-16 (row = t>>4, columns 8*(t&15) .. 8*(t&15)+7)
    float bd[KNN]; int bi[KNN];
    #pragma unroll
    for (int i = 0; i < KNN; ++i) { bd[i] = 3.0e38f; bi[i] = 0x7fffffff; }
    const int selRow = t >> 4, selSub = t & 15;

    const int nB = wave * 16 + (lane & 15);
    const int koffB = (lane < 16) ? 0 : 2;
    const int mofs = (lane < 16) ? 0 : 8;

    for (int s = 0; s < nsteps; ++s) {
        const int colbase = cbase + s * STEP;
        __syncthreads();                     // buffer (s+1)&1 consumption (step s-1) done
        if (wave == 0) {
            if (s + 1 < nsteps) {            // prefetch next tile, keep it in flight
                tdm_load_tile((uint32_t)(uintptr_t)&xbuf[(s + 1) & 1][0],
                              X + (size_t)(colbase + STEP) * DIM, N - (colbase + STEP));
                __builtin_amdgcn_s_wait_tensorcnt(1);
            } else {
                __builtin_amdgcn_s_wait_tensorcnt(0);
            }
        }
        __syncthreads();                     // xbuf[s&1] visible to all waves

        const float* xs = &xbuf[s & 1][0];
        v8f acc = {};
        #pragma unroll
        for (int kb = 0; kb < 32; ++kb) {
            v2f bfrag = *(const v2f*)&xs[nB * XPAD + kb * 4 + koffB];
            acc = __builtin_amdgcn_wmma_f32_16x16x4_f32(
                false, afrag[kb], false, bfrag, (short)0, acc, false, false);
        }
        // d2 (minus per-row const) = ||x||^2 - 2*dot ; OOB rows -> +inf
        {
            float xnn = (colbase + nB < N) ? xnorm[colbase + nB] : 3.0e38f;
            #pragma unroll
            for (int r = 0; r < 8; ++r)
                dist[(r + mofs) * XPAD + nB] = fmaf(-2.0f, acc[r], xnn);
        }
        __syncthreads();

        // selection: each thread scans 8 columns of its row (register insertion sort)
        #pragma unroll
        for (int j = 0; j < 8; ++j) {
            int c = selSub * 8 + j;
            float dv = dist[selRow * XPAD + c];
            int gi = colbase + c;
            if (dv < bd[KNN - 1]) {
                int pos = KNN - 1;
                #pragma unroll
                for (int u = KNN - 2; u >= 0; --u)
                    if (dv < bd[u]) { bd[u + 1] = bd[u]; bi[u + 1] = bi[u]; pos = u; }
                bd[pos] = dv; bi[pos] = gi;
            }
        }
    }
    __syncthreads();                          // all TDM done (wait 0 in last iter) + reads done

    // dump 16 rows x 256 candidates into xbuf[0], merge 256 -> 16 per row
    float* cD = &xbuf[0][0];
    int*   cI = (int*)&xbuf[0][16 * 256];
    #pragma unroll
    for (int m = 0; m < KNN; ++m) {
        cD[selRow * 256 + selSub * 16 + m] = bd[m];
        cI[selRow * 256 + selSub * 16 + m] = bi[m];
    }
    __syncthreads();

    const int nc16 = nch * 16;
    for (int rr = wave; rr < 16; rr += 8) {
        float ld[8]; int li[8];
        #pragma unroll
        for (int j = 0; j < 8; ++j) {
            int p = j * 32 + lane;
            ld[j] = cD[rr * 256 + p];
            li[j] = cI[rr * 256 + p];
        }
        int rowg = rg * 16 + rr;
        for (int m = 0; m < KNN; ++m) {
            float mv = 3.0e38f; int mi = 0x7fffffff;
            #pragma unroll
            for (int j = 0; j < 8; ++j)
                if (ld[j] < mv || (ld[j] == mv && li[j] < mi)) { mv = ld[j]; mi = li[j]; }
            float v = mv; int idx = mi;
            #pragma unroll
            for (int off = 16; off; off >>= 1) {
                float ov = __shfl_xor(v, off, 32);
                int   oi = __shfl_xor(idx, off, 32);
                if (ov < v || (ov == v && oi < idx)) { v = ov; idx = oi; }
            }
            if (lane == 0) {
                candD[(size_t)rowg * nc16 + ch * 16 + m] = v;
                candI[(size_t)rowg * nc16 + ch * 16 + m] = idx;
            }
            // static-index removal: (v, idx) unique among real candidates
            #pragma unroll
            for (int j = 0; j < 8; ++j)
                if (ld[j] == v && li[j] == idx) { ld[j] = 3.0e38f; li[j] = 0x7fffffff; }
        }
    }
}

// ----- Kernel 3: global top-16 merge + exact l2 + softmax + union KL ------
__global__ __launch_bounds__(32) void knn_finalize(const float* __restrict__ Tq,
                                                   const float* __restrict__ X,
                                                   const int* __restrict__ q_indices,
                                                   const int* __restrict__ pre_idx,
                                                   const float* __restrict__ pre_w,
                                                   const float* __restrict__ candD,
                                                   const int* __restrict__ candI,
                                                   int nch, float* __restrict__ klb) {
    const int b = blockIdx.x;
    const int lane = threadIdx.x;
    const int ncand = nch * 16;                 // <= 1024

    float cd[32]; int ci[32];
    #pragma unroll
    for (int j = 0; j < 32; ++j) {
        int p = j * 32 + lane;
        if (p < ncand) { cd[j] = candD[(size_t)b * ncand + p]; ci[j] = candI[(size_t)b * ncand + p]; }
        else           { cd[j] = 3.0e38f;       ci[j] = 0x7fffffff; }
    }
    __shared__ int nb[KNN];
    for (int m = 0; m < KNN; ++m) {
        float mv = 3.0e38f; int mi = 0x7fffffff;
        #pragma unroll
        for (int j = 0; j < 32; ++j)
            if (cd[j] < mv || (cd[j] == mv && ci[j] < mi)) { mv = cd[j]; mi = ci[j]; }
        float v = mv; int idx = mi;
        #pragma unroll
        for (int off = 16; off; off >>= 1) {
            float ov = __shfl_xor(v, off, 32);
            int   oi = __shfl_xor(idx, off, 32);
            if (ov < v || (ov == v && oi < idx)) { v = ov; idx = oi; }
        }
        if (lane == 0) nb[m] = idx;
        #pragma unroll
        for (int j = 0; j < 32; ++j)
            if (cd[j] == v && ci[j] == idx) { cd[j] = 3.0e38f; ci[j] = 0x7fffffff; }
    }
    __syncthreads();

    // exact fp32 l2 to the 16 neighbors: lane = (neighbor j = lane&15, half = lane>>4)
    const int j = lane & 15, half = lane >> 4;
    const int nidx = nb[j];
    const float* xr = X  + (size_t)nidx * DIM + half * 64;
    const float* tr = Tq + (size_t)b    * DIM + half * 64;
    float s2 = 0.f;
    #pragma unroll 8
    for (int k = 0; k < 64; ++k) { float d = tr[k] - xr[k]; s2 = fmaf(d, d, s2); }
    s2 += __shfl_xor(s2, 16, 32);               // lanes j and j+16 both get full l2

    // softmax(-l2/tau) with max subtraction (groups of 16 replicated)
    float mn = s2;
    #pragma unroll
    for (int off = 8; off; off >>= 1) mn = fminf(mn, __shfl_xor(mn, off, 32));
    float w = expf((mn - s2) / TAU);
    float sw = w;
    #pragma unroll
    for (int off = 8; off; off >>= 1) sw += __shfl_xor(sw, off, 32);
    float qw = w / sw;

    __shared__ float qws[KNN];
    if (lane < KNN) qws[lane] = qw;
    __syncthreads();

    if (lane == 0) {
        int ui[2 * KNN]; float pv[2 * KNN]; float qv[2 * KNN]; int cnt = KNN;
        for (int u = 0; u < KNN; ++u) { ui[u] = nb[u]; qv[u] = qws[u]; pv[u] = 0.f; }
        int qi = q_indices[b];
        for (int u = 0; u < KNN; ++u) {         // jax scatter: duplicates -> last wins
            int idx = pre_idx[qi * KNN + u]; float wv = pre_w[qi * KNN + u];
            int pos = -1;
            for (int x = 0; x < cnt; ++x) if (ui[x] == idx) { pos = x; break; }
            if (pos < 0) { ui[cnt] = idx; pv[cnt] = wv; qv[cnt] = 0.f; ++cnt; }
            else pv[pos] = wv;
        }
        float ps = 0.f, qs = 0.f;
        for (int u = 0; u < cnt; ++u) {
            bool in = (pv[u] > 0.f) || (qv[u] > 0.f);
            if (in) { pv[u] = fmaxf(pv[u], EPSF); qv[u] = fmaxf(qv[u], EPSF); ps += pv[u]; qs += qv[u]; }
            else    { pv[u] = 0.f; qv[u] = 0.f; }
        }
        float kl = 0.f;
        for (int u = 0; u < cnt; ++u)
            if (pv[u] > 0.f) {
                float p = pv[u] / ps, q2 = qv[u] / qs;
                kl += p * (logf(fmaxf(p, 1e-20f)) - logf(fmaxf(q2, 1e-20f)));
            }
        klb[b] = kl;
    }
}

// ---------------- Kernel 4: final deterministic reduction -----------------
__global__ __launch_bounds__(256) void finalize_out(const float* __restrict__ klb,
                                                    const float* __restrict__ W,
                                                    float* __restrict__ out,
                                                    int Bn, int wlen) {
    __shared__ float sk[256];
    __shared__ float sr[256];
    int t = threadIdx.x;
    float a = 0.f, r = 0.f;
    for (int i = t; i < Bn; i += 256) a += klb[i];
    for (int i = t; i < wlen; i += 256) { float w = W[i]; r = fmaf(w, w, r); }
    sk[t] = a; sr[t] = r;
    __syncthreads();
    for (int o = 128; o; o >>= 1) {
        if (t < o) { sk[t] += sk[t + o]; sr[t] += sr[t + o]; }
        __syncthreads();
    }
    if (t == 0) {
        float loss_knn = sk[0] / (float)Bn;
        float loss_reg = 0.5f * sr[0];
        out[0] = BETA * loss_knn + LAMB * loss_reg;   // total_loss
        out[1] = 0.0f;                                // loss_dist
        out[2] = loss_knn;                            // loss_knn
    }
}

extern "C" void kernel_launch(void* const* d_in, const int* in_sizes, int n_in,
                              void* d_out, int out_size, void* d_ws, size_t ws_size,
                              hipStream_t stream) {
    const float* q_batch   = (const float*)d_in[0];
    const int*   q_indices = (const int*)  d_in[1];
    const float* X         = (const float*)d_in[2];
    const float* W         = (const float*)d_in[3];
    const int*   pre_idx   = (const int*)  d_in[4];
    const float* pre_w     = (const float*)d_in[5];

    const int Bn   = in_sizes[1];           // 256
    const int N    = in_sizes[2] / DIM;     // 100000
    const int wlen = in_sizes[3];           // 128*128
    const int nch  = (N + CHUNK - 1) / CHUNK;

    // workspace layout
    float* Tq    = (float*)d_ws;                          // Bn*DIM
    float* xnorm = Tq + (size_t)Bn * DIM;                 // N
    float* candD = xnorm + (size_t)N;                     // Bn*nch*16
    int*   candI = (int*)(candD + (size_t)Bn * nch * 16); // Bn*nch*16
    float* klb   = (float*)(candI + (size_t)Bn * nch * 16);

    tq_project<<<Bn, 128, 0, stream>>>(q_batch, W, Tq);
    x_norms<<<(N + 255) / 256, 256, 0, stream>>>(X, xnorm, N);

    dim3 gB(Bn / 16, nch);
    knn_dist_topk<<<gB, 256, 0, stream>>>(Tq, X, xnorm, N, nch, candD, candI);

    knn_finalize<<<Bn, 32, 0, stream>>>(Tq, X, q_indices, pre_idx, pre_w,
                                        candD, candI, nch, klb);

    finalize_out<<<1, 256, 0, stream>>>(klb, W, (float*)d_out, Bn, wlen);
}